// TPS_20392504721814
// MI455X (gfx1250) — compile-verified
//
#include <hip/hip_runtime.h>
#include <hip/hip_bf16.h>

// ---------------------------------------------------------------------------
// TPS (thin-plate spline) forward on gfx1250 / MI455X.
//   in0: source_points (1,65536,3) f32
//   in1: points        (1,4098,3)  f32   (control points, "ctrl")
//   in2: sphere_points (1,4098,3)  f32   (targets, "tgt")
//   out: (1,65536,3) f32
//
// Pipeline:
//   1) build padded augmented system  A(4128 x 4144) (RHS in cols 4128..4131)
//   2) blocked LU w/ partial pivoting (panel=32):
//        panel_factor (1 block) -> trsm (grid) ->
//        trailing update: TDM (tensor_load_to_lds) stages L21/U12 into LDS,
//        8 waves/block run V_WMMA_F32_16X16X4_F32 from ds_load operands.
//   3) back-substitution -> zero-padded RBF coef table + tiny affine table
//   4) fused eval: r^2 tile via V_WMMA_F32_16X16X4_F32, U = .5 r2 log r2,
//      accumulate against coef, cross-lane reduce, add affine part.
// ---------------------------------------------------------------------------

typedef float v2f __attribute__((ext_vector_type(2)));
typedef float v8f __attribute__((ext_vector_type(8)));
typedef unsigned int v4u __attribute__((ext_vector_type(4)));
typedef int v4i __attribute__((ext_vector_type(4)));
typedef int v8i __attribute__((ext_vector_type(8)));

#define NSRC   65536
#define P      4098          // control points
#define NA     (P + 4)       // 4102 real system size
#define NPAD   4128          // system padded to multiple of 32 (129 panels)
#define LDA    (NPAD + 16)   // 4 RHS cols + 12 slack cols for 16-wide tiles
#define PPAD   4112          // P padded to multiple of 16 (257 chunks)
#define PANEL  32
#define SUPER  128           // trailing-update super-tile (128x128 per block)
#define EPSR   1e-18f

__device__ __forceinline__ float tps_u(float r2) {
  r2 = fmaxf(r2, 0.0f);
  return 0.5f * r2 * __logf(fmaxf(r2, EPSR));
}

// Low 32 bits of a generic pointer to LDS == DS-instruction byte offset
// (flat->LDS lowering truncates the address the same way; ISA 10.2).
__device__ __forceinline__ unsigned lds_off(const void* p) {
  return (unsigned)(unsigned long long)p;
}

// --- Tensor DMA descriptor packing (CDNA5 ISA ch.8, D# group 0/1) ----------
__device__ __forceinline__ v4u tdm_g0(unsigned lds_addr, unsigned long long gaddr) {
  v4u g;
  g[0] = 1u;                                   // count=1, user mode, no gather
  g[1] = lds_addr;                             // LDS byte address
  g[2] = (unsigned)gaddr;                      // global addr [31:0]
  g[3] = (unsigned)((gaddr >> 32) & 0x01FFFFFFull) | (2u << 30); // [56:32]|type=2
  return g;
}
__device__ __forceinline__ v8i tdm_g1(unsigned tensor_d0, unsigned tensor_d1,
                                      unsigned tile_d0, unsigned tile_d1,
                                      unsigned long long stride0_elems) {
  v8i g;
  g[0] = (int)(2u << 16);                                      // data_size=4B
  g[1] = (int)((tensor_d0 & 0xFFFFu) << 16);                   // dim0[15:0]
  g[2] = (int)((tensor_d0 >> 16) | ((tensor_d1 & 0xFFFFu) << 16));
  g[3] = (int)((tensor_d1 >> 16) | (tile_d0 << 16));           // tile_dim0
  g[4] = (int)tile_d1;                                         // tile_dim1 (dim2=0)
  g[5] = (int)(unsigned)stride0_elems;                         // stride0[31:0]
  g[6] = (int)(unsigned)((stride0_elems >> 32) & 0xFFFFull);   // stride0[47:32]
  g[7] = 0;                                                    // stride1=0 (2D)
  return g;
}

// ---------------------------------------------------------------------------
// 1) Build padded augmented matrix. Padding region is identity, RHS rows >= P
//    are zero, so the padded solve equals the real solve on rows 0..NA-1.
// ---------------------------------------------------------------------------
__global__ __launch_bounds__(256) void build_A(const float* __restrict__ ctrl,
                                               const float* __restrict__ tgt,
                                               float* __restrict__ A) {
  long long idx = (long long)blockIdx.x * blockDim.x + threadIdx.x;
  if (idx >= (long long)NPAD * LDA) return;
  int i = (int)(idx / LDA), j = (int)(idx % LDA);
  float val = 0.0f;
  if (j < NPAD) {
    if (i < P && j < P) {
      float dx = ctrl[i * 3 + 0] - ctrl[j * 3 + 0];
      float dy = ctrl[i * 3 + 1] - ctrl[j * 3 + 1];
      float dz = ctrl[i * 3 + 2] - ctrl[j * 3 + 2];
      val = tps_u(dx * dx + dy * dy + dz * dz);     // diag = U(0) = 0
    } else if (i < P && j >= P && j < NA) {
      val = (j == P) ? 1.0f : ctrl[i * 3 + (j - P - 1)];
    } else if (i >= P && i < NA && j < P) {
      val = (i == P) ? 1.0f : ctrl[j * 3 + (i - P - 1)];
    } else if (i >= NA || j >= NA) {
      val = (i == j) ? 1.0f : 0.0f;                 // identity padding
    }                                               // [P,NA)x[P,NA) block = 0
  } else {
    int d = j - NPAD;
    val = (i < P && d < 3) ? tgt[i * 3 + d] : 0.0f; // RHS (+1 zero col)
  }
  A[idx] = val;
}

// ---------------------------------------------------------------------------
// 2a) Panel factorization: 32 columns, partial pivoting over full column,
//     full-width row swaps (covers RHS too), one 1024-thread block.
// ---------------------------------------------------------------------------
__global__ __launch_bounds__(1024) void panel_factor(float* __restrict__ A, int c0) {
  __shared__ float sval[1024];
  __shared__ int   sidx[1024];
  const int tid = threadIdx.x;
  for (int jj = 0; jj < PANEL; ++jj) {
    const int j = c0 + jj;
    float best = -1.0f; int bidx = j;
    for (int r = j + tid; r < NPAD; r += 1024) {
      float v = fabsf(A[(long long)r * LDA + j]);
      if (v > best) { best = v; bidx = r; }
    }
    sval[tid] = best; sidx[tid] = bidx;
    __syncthreads();
    for (int s = 512; s > 0; s >>= 1) {
      if (tid < s && sval[tid + s] > sval[tid]) {
        sval[tid] = sval[tid + s]; sidx[tid] = sidx[tid + s];
      }
      __syncthreads();
    }
    const int pr = sidx[0];
    __syncthreads();
    if (pr != j) {
      for (int c = tid; c < LDA; c += 1024) {
        float t = A[(long long)j * LDA + c];
        A[(long long)j * LDA + c]  = A[(long long)pr * LDA + c];
        A[(long long)pr * LDA + c] = t;
      }
    }
    __syncthreads();
    const float inv = 1.0f / A[(long long)j * LDA + j];
    for (int r = j + 1 + tid; r < NPAD; r += 1024) {
      long long ro = (long long)r * LDA;
      float l = A[ro + j] * inv;
      A[ro + j] = l;
      for (int c = jj + 1; c < PANEL; ++c)
        A[ro + c0 + c] -= l * A[(long long)j * LDA + c0 + c];
    }
    __syncthreads();
  }
}

// ---------------------------------------------------------------------------
// 2b) TRSM: A[c0:c0+32, c0+32:LDA] = L11^{-1} * strip (unit lower L11).
// ---------------------------------------------------------------------------
__global__ __launch_bounds__(256) void trsm_u12(float* __restrict__ A, int c0) {
  int j = c0 + PANEL + blockIdx.x * blockDim.x + threadIdx.x;
  if (j >= LDA) return;
  float x[PANEL];
#pragma unroll 4
  for (int r = 0; r < PANEL; ++r) {
    long long ro = (long long)(c0 + r) * LDA;
    float v = A[ro + j];
    for (int t = 0; t < r; ++t) v -= A[ro + c0 + t] * x[t];
    x[r] = v;
    A[ro + j] = v;
  }
}

// ---------------------------------------------------------------------------
// 2c) Trailing update  A22 -= L21 * U12.
//     Per block: 128x128 super-tile. Wave 0 issues two Tensor-Data-Mover
//     loads (U12 32x128 strip, L21 128x32 block) into LDS, waits on
//     TENSORcnt, then 8 waves compute 16x128 strips with chained
//     V_WMMA_F32_16X16X4_F32. Both operands fully preloaded into VGPRs per
//     tile so the DS clause waits once and the 8 WMMAs issue back-to-back.
//     TDM tensor_dim is clipped to the valid extent so overhang zero-fills.
// ---------------------------------------------------------------------------
__global__ __launch_bounds__(256) void lu_gemm_wmma(float* __restrict__ A, int c0,
                                                    int colEnd, int nSCt) {
  __shared__ float sU[PANEL * SUPER];   // U12: 32 rows x 128 cols  (16 KB)
  __shared__ float sL[SUPER * PANEL];   // L21: 128 rows x 32 cols  (16 KB)

  const int base = c0 + PANEL;
  const int sr = blockIdx.x / nSCt, sc = blockIdx.x % nSCt;
  const int R0 = base + sr * SUPER;
  const int C0 = base + sc * SUPER;
  const int rowsAvail = NPAD - R0;              // > 0, multiple of 32
  const int colsAvail = colEnd - C0;            // > 0, multiple of 16

  if (threadIdx.x < 32) {                       // wave 0 drives the TDM
    unsigned long long ab = (unsigned long long)A;
    v4i z4 = {0, 0, 0, 0};
    v8i z8 = {0, 0, 0, 0, 0, 0, 0, 0};
    // U12 strip: rows c0..c0+31, cols C0..C0+127 (zero-fill past colsAvail)
    v4u g0u = tdm_g0(lds_off(sU), ab + ((unsigned long long)c0 * LDA + C0) * 4ull);
    v8i g1u = tdm_g1((unsigned)colsAvail, PANEL, SUPER, PANEL, LDA);
    __builtin_amdgcn_tensor_load_to_lds(g0u, g1u, z4, z4, z8, 0);
    // L21 block: rows R0..R0+127 (zero-fill past rowsAvail), cols c0..c0+31
    v4u g0l = tdm_g0(lds_off(sL), ab + ((unsigned long long)R0 * LDA + c0) * 4ull);
    v8i g1l = tdm_g1(PANEL, (unsigned)rowsAvail, PANEL, SUPER, LDA);
    __builtin_amdgcn_tensor_load_to_lds(g0l, g1l, z4, z4, z8, 0);
    __builtin_amdgcn_s_wait_tensorcnt(0);
  }
  __syncthreads();

  const int wv = threadIdx.x >> 5;
  const int lane = threadIdx.x & 31;
  const int half = lane >> 4, lr = lane & 15;

  if (wv * 16 < rowsAvail) {                    // wave-uniform guard
    const int lrow = wv * 16 + lr;
    // Preload negated A-operand (L21) once per wave: 8 K-steps x 2 VGPRs.
    v2f areg[8];
#pragma unroll
    for (int k8 = 0; k8 < 8; ++k8) {
      const float* ap = &sL[lrow * PANEL + k8 * 4 + half * 2];
      areg[k8][0] = -ap[0];
      areg[k8][1] = -ap[1];
    }
    const int gr0 = R0 + wv * 16;
#pragma unroll 1
    for (int ct = 0; ct < SUPER / 16; ++ct) {
      const int n0 = C0 + ct * 16;
      if (n0 >= colEnd) break;                  // uniform
      // Preload the full B-operand column tile (one DS clause, one wait).
      v2f breg[8];
#pragma unroll
      for (int k8 = 0; k8 < 8; ++k8) {
        breg[k8][0] = sU[(k8 * 4 + half * 2 + 0) * SUPER + ct * 16 + lr];
        breg[k8][1] = sU[(k8 * 4 + half * 2 + 1) * SUPER + ct * 16 + lr];
      }
      v8f c;
#pragma unroll
      for (int v = 0; v < 8; ++v)
        c[v] = A[(long long)(gr0 + v + half * 8) * LDA + n0 + lr];
#pragma unroll
      for (int k8 = 0; k8 < 8; ++k8)
        c = __builtin_amdgcn_wmma_f32_16x16x4_f32(false, areg[k8], false, breg[k8],
                                                  (short)0, c, false, false);
#pragma unroll
      for (int v = 0; v < 8; ++v)
        A[(long long)(gr0 + v + half * 8) * LDA + n0 + lr] = c[v];
    }
  }
}

// ---------------------------------------------------------------------------
// 3) Back-substitution (U x = y, 3 RHS). RBF coef table zero-padded past P
//    (so eval needs no masking); affine rows P..P+3 land in aff[4][4].
// ---------------------------------------------------------------------------
__global__ __launch_bounds__(1024) void backsubst(float* __restrict__ A,
                                                  float* __restrict__ coef,
                                                  float* __restrict__ aff) {
  __shared__ float xs[3];
  const int tid = threadIdx.x;
  for (int i = NPAD - 1; i >= 0; --i) {
    long long io = (long long)i * LDA;
    if (tid < 3) xs[tid] = A[io + NPAD + tid] / A[io + i];
    __syncthreads();
    if (i < PPAD && tid < 4)
      coef[i * 4 + tid] = (i < P && tid < 3) ? xs[tid] : 0.0f;
    if (i >= P && i < NA && tid < 4)
      aff[(i - P) * 4 + tid] = (tid < 3) ? xs[tid] : 0.0f;
    for (int r = tid; r < i; r += 1024) {
      long long ro = (long long)r * LDA;
      float u = A[ro + i];
      A[ro + NPAD + 0] -= u * xs[0];
      A[ro + NPAD + 1] -= u * xs[1];
      A[ro + NPAD + 2] -= u * xs[2];
    }
    __syncthreads();
  }
}

// ---------------------------------------------------------------------------
// 4a) Pack B-operand per control point: (-2x,-2y,-2z,|c|^2); zeros in padding
//     so padded columns contribute r2 = |s|^2 with coef already zero there.
// ---------------------------------------------------------------------------
__global__ __launch_bounds__(256) void prep_bpack(const float* __restrict__ ctrl,
                                                  float* __restrict__ bpack) {
  int k = blockIdx.x * blockDim.x + threadIdx.x;
  if (k >= PPAD) return;
  float x = 0.f, y = 0.f, z = 0.f, c2 = 0.f;
  if (k < P) {
    x = ctrl[k * 3 + 0]; y = ctrl[k * 3 + 1]; z = ctrl[k * 3 + 2];
    c2 = x * x + y * y + z * z;
  }
  bpack[k * 4 + 0] = -2.0f * x;
  bpack[k * 4 + 1] = -2.0f * y;
  bpack[k * 4 + 2] = -2.0f * z;
  bpack[k * 4 + 3] = c2;
}

// ---------------------------------------------------------------------------
// 4b) Fused evaluation. One wave = 16 query rows. Per 16-control chunk:
//     r2 tile = WMMA(A=[x,y,z,1], B=[-2c | |c|^2], C=|s|^2 bcast), then
//     U = .5 r2 log r2 and FMA against per-lane coef; final 16-lane reduce.
// ---------------------------------------------------------------------------
__global__ __launch_bounds__(256) void tps_eval(const float* __restrict__ src,
                                               const float* __restrict__ bpack,
                                               const float* __restrict__ coef,
                                               const float* __restrict__ aff,
                                               float* __restrict__ out) {
  const int wave = threadIdx.x >> 5;
  const int lane = threadIdx.x & 31;
  const int m0 = (blockIdx.x * (blockDim.x >> 5) + wave) * 16;
  const int half = lane >> 4, lr = lane & 15;

  const float px = src[(m0 + lr) * 3 + 0];
  const float py = src[(m0 + lr) * 3 + 1];
  const float pz = src[(m0 + lr) * 3 + 2];
  const float a2 = px * px + py * py + pz * pz;

  v2f amat;                       // A 16x4: [x,y | z,1]
  amat[0] = half ? pz : px;
  amat[1] = half ? 1.0f : py;

  v8f c0v;                        // C-layout broadcast of |s|^2 + row coords
  float xs[8], ys[8], zs[8];
#pragma unroll
  for (int v = 0; v < 8; ++v) {
    int sl = v + half * 8;
    c0v[v] = __shfl(a2, sl, 32);
    xs[v]  = __shfl(px, sl, 32);
    ys[v]  = __shfl(py, sl, 32);
    zs[v]  = __shfl(pz, sl, 32);
  }

  float acc[8][3] = {};
#pragma unroll 1
  for (int j0 = 0; j0 < PPAD; j0 += 16) {
    const int k = j0 + lr;
    const float2 bb = *(const float2*)&bpack[k * 4 + half * 2];
    const float4 cf = *(const float4*)&coef[k * 4];     // zero for k >= P
    v2f b; b[0] = bb.x; b[1] = bb.y;
    v8f r2 = __builtin_amdgcn_wmma_f32_16x16x4_f32(false, amat, false, b,
                                                   (short)0, c0v, false, false);
#pragma unroll
    for (int v = 0; v < 8; ++v) {
      float r = fmaxf(r2[v], 0.0f);
      float u = 0.5f * r * __logf(fmaxf(r, EPSR));
      acc[v][0] = fmaf(u, cf.x, acc[v][0]);
      acc[v][1] = fmaf(u, cf.y, acc[v][1]);
      acc[v][2] = fmaf(u, cf.z, acc[v][2]);
    }
  }

#pragma unroll
  for (int off = 1; off < 16; off <<= 1)
#pragma unroll
    for (int v = 0; v < 8; ++v) {
      acc[v][0] += __shfl_xor(acc[v][0], off, 32);
      acc[v][1] += __shfl_xor(acc[v][1], off, 32);
      acc[v][2] += __shfl_xor(acc[v][2], off, 32);
    }

  if (lr == 0) {                    // lanes 0 and 16 hold rows v / v+8
#pragma unroll
    for (int v = 0; v < 8; ++v) {
      int m = v + half * 8;
#pragma unroll
      for (int d = 0; d < 3; ++d) {
        float val = acc[v][d]
                  + aff[0 * 4 + d]
                  + xs[v] * aff[1 * 4 + d]
                  + ys[v] * aff[2 * 4 + d]
                  + zs[v] * aff[3 * 4 + d];
        out[(m0 + m) * 3 + d] = val;
      }
    }
  }
}

// ---------------------------------------------------------------------------
extern "C" void kernel_launch(void* const* d_in, const int* in_sizes, int n_in,
                              void* d_out, int out_size, void* d_ws, size_t ws_size,
                              hipStream_t stream) {
  (void)in_sizes; (void)n_in; (void)out_size; (void)ws_size;
  const float* src  = (const float*)d_in[0];   // (1,65536,3)
  const float* ctrl = (const float*)d_in[1];   // (1,4098,3)
  const float* tgt  = (const float*)d_in[2];   // (1,4098,3)
  float* out = (float*)d_out;

  float* A     = (float*)d_ws;                          // NPAD*LDA  (~68 MB)
  float* bpack = A + (long long)NPAD * LDA;             // PPAD*4
  float* coefp = bpack + (long long)PPAD * 4;           // PPAD*4
  float* affp  = coefp + (long long)PPAD * 4;           // 16 floats

  {
    long long tot = (long long)NPAD * LDA;
    build_A<<<(unsigned)((tot + 255) / 256), 256, 0, stream>>>(ctrl, tgt, A);
  }
  prep_bpack<<<(PPAD + 255) / 256, 256, 0, stream>>>(ctrl, bpack);

  for (int c0 = 0; c0 < NPAD; c0 += PANEL) {
    panel_factor<<<1, 1024, 0, stream>>>(A, c0);
    int ncols = LDA - (c0 + PANEL);
    if (ncols > 0)
      trsm_u12<<<(ncols + 255) / 256, 256, 0, stream>>>(A, c0);
    int base = c0 + PANEL;
    int rows = NPAD - base;
    if (rows > 0) {
      int colEnd = base + (((NPAD + 4) - base + 15) / 16) * 16;  // <= LDA
      int nSRt = (rows + SUPER - 1) / SUPER;
      int nSCt = (colEnd - base + SUPER - 1) / SUPER;
      lu_gemm_wmma<<<nSRt * nSCt, 256, 0, stream>>>(A, c0, colEnd, nSCt);
    }
  }

  backsubst<<<1, 1024, 0, stream>>>(A, coefp, affp);

  // 65536 rows / (16 rows/wave * 8 waves/block) = 512 blocks, no remainder
  tps_eval<<<NSRC / 128, 256, 0, stream>>>(src, bpack, coefp, affp, out);
}